// LEIRSMPL_26697516712143
// MI455X (gfx1250) — compile-verified
//
#include <hip/hip_runtime.h>
#include <hip/hip_bf16.h>

// ---------------- SMPL constants ----------------
#define NV 6890
#define NJ 24
#define NB 1024
#define NP 207            // pose blendshape K
#define KS 10             // shape K
#define KTOT 220          // 207 pose + 1 pad + 10 shape + 2 pad
#define M_ROWS 20670      // NV*3
#define TILE_V 32         // vertices per tile -> 96 M rows
#define TILE_B 32         // bodies per tile
#define N_VT   216        // ceil(20670 / 96)
#define M_PAD  (N_VT*96)  // 20736

// workspace layout (float offsets)
#define WS_JSJT 0                        // 24*33
#define WS_B    1024                     // NB*220   packed B (lrotmin|beta, zero-padded)
#define WS_R    (WS_B + NB*KTOT)         // NB*24*9
#define WS_J    (WS_R + NB*NJ*9)         // NB*24*3
#define WS_G    (WS_J + NB*NJ*3)         // NB*24*12
#define WS_A    (WS_G + NB*NJ*12)        // M_PAD*220 packed A (posedirs|shapedirs, zero-padded)

__constant__ int c_parent[23] = {0,0,0,1,2,3,4,5,6,7,8,9,9,9,12,13,14,16,17,18,19,20,21};

typedef __attribute__((ext_vector_type(2))) float v2f;
typedef __attribute__((ext_vector_type(8))) float v8f;

// ---------------- K0: pack A = [posedirs | 0 | shapedirs | 0 0], rows padded to M_PAD ----------------
__global__ __launch_bounds__(256) void k_pack_A(
    const float* __restrict__ posedirs, const float* __restrict__ shapedirs,
    float* __restrict__ Apack) {
  int idx = blockIdx.x * blockDim.x + threadIdx.x;
  if (idx >= M_PAD * KTOT) return;
  int R = idx / KTOT, k = idx - R * KTOT;
  float v = 0.f;
  if (R < M_ROWS) {
    if (k < NP)                   v = posedirs[(size_t)R * NP + k];
    else if (k >= 208 && k < 218) v = shapedirs[(size_t)R * KS + (k - 208)];
  }
  Apack[idx] = v;
}

// ---------------- K1: fold J_regressor into shapedirs/template ----------------
__global__ __launch_bounds__(128) void k_fold_regressor(
    const float* __restrict__ Jreg, const float* __restrict__ shapedirs,
    const float* __restrict__ v_template, float* __restrict__ jsjt) {
  int j = blockIdx.x, tid = threadIdx.x;
  float part[33];
#pragma unroll
  for (int o = 0; o < 33; ++o) part[o] = 0.f;
  for (int v = tid; v < NV; v += 128) {
    float jr = Jreg[j*NV + v];
#pragma unroll
    for (int c = 0; c < 3; ++c) {
#pragma unroll
      for (int s = 0; s < 10; ++s)
        part[c*10+s] += jr * shapedirs[v*30 + c*10 + s];
      part[30+c] += jr * v_template[v*3 + c];
    }
  }
  __shared__ float red[128];
  for (int o = 0; o < 33; ++o) {
    red[tid] = part[o];
    __syncthreads();
    for (int off = 64; off > 0; off >>= 1) {
      if (tid < off) red[tid] += red[tid + off];
      __syncthreads();
    }
    if (tid == 0) jsjt[j*33 + o] = red[0];
    __syncthreads();
  }
}

// ---------------- K2a: Rodrigues -> R, (R-I) -> Bpad, joints ----------------
__global__ __launch_bounds__(256) void k_pose_prep(
    const float* __restrict__ pose, const float* __restrict__ beta,
    const float* __restrict__ jsjt, float* __restrict__ wsR,
    float* __restrict__ wsJ, float* __restrict__ Bpad) {
  int idx = blockIdx.x * blockDim.x + threadIdx.x;
  if (idx >= NB * NJ) return;
  int b = idx / NJ, j = idx % NJ;
  float rx = pose[b*72 + j*3 + 0];
  float ry = pose[b*72 + j*3 + 1];
  float rz = pose[b*72 + j*3 + 2];
  float ang = sqrtf(rx*rx + ry*ry + rz*rz + 1e-8f);
  float inv = 1.0f / ang;
  float x = rx*inv, y = ry*inv, z = rz*inv;
  float c = cosf(ang), s = sinf(ang), t = 1.0f - c;
  float R[9];
  R[0] = c + t*x*x;   R[1] = t*x*y - s*z; R[2] = t*x*z + s*y;
  R[3] = t*x*y + s*z; R[4] = c + t*y*y;   R[5] = t*y*z - s*x;
  R[6] = t*x*z - s*y; R[7] = t*y*z + s*x; R[8] = c + t*z*z;
#pragma unroll
  for (int e = 0; e < 9; ++e) wsR[(b*NJ + j)*9 + e] = R[e];
  if (j >= 1) {
    int base = b*KTOT + (j-1)*9;
#pragma unroll
    for (int e = 0; e < 9; ++e)
      Bpad[base + e] = R[e] - ((e % 4 == 0) ? 1.0f : 0.0f);
  } else {
    Bpad[b*KTOT + 207] = 0.0f;
#pragma unroll
    for (int s2 = 0; s2 < 10; ++s2) Bpad[b*KTOT + 208 + s2] = beta[b*10 + s2];
    Bpad[b*KTOT + 218] = 0.0f;
    Bpad[b*KTOT + 219] = 0.0f;
  }
#pragma unroll
  for (int cc = 0; cc < 3; ++cc) {
    float acc = jsjt[j*33 + 30 + cc];
#pragma unroll
    for (int ss = 0; ss < 10; ++ss)
      acc += jsjt[j*33 + cc*10 + ss] * beta[b*10 + ss];
    wsJ[(b*NJ + j)*3 + cc] = acc;
  }
}

// ---------------- K2b: forward kinematics + rest-pose correction ----------------
__global__ __launch_bounds__(64) void k_fk(
    const float* __restrict__ wsR, const float* __restrict__ wsJ,
    float* __restrict__ wsG) {
  int b = blockIdx.x * blockDim.x + threadIdx.x;
  if (b >= NB) return;
  float Aw[NJ][9], tw[NJ][3], Jl[NJ][3];
  for (int j = 0; j < NJ; ++j)
    for (int cc = 0; cc < 3; ++cc) Jl[j][cc] = wsJ[(b*NJ + j)*3 + cc];
  for (int e = 0; e < 9; ++e) Aw[0][e] = wsR[(b*NJ)*9 + e];
  for (int cc = 0; cc < 3; ++cc) tw[0][cc] = Jl[0][cc];
  for (int j = 1; j < NJ; ++j) {
    int p = c_parent[j-1];
    float Rl[9];
    for (int e = 0; e < 9; ++e) Rl[e] = wsR[(b*NJ + j)*9 + e];
    float tl[3];
    for (int cc = 0; cc < 3; ++cc) tl[cc] = Jl[j][cc] - Jl[p][cc];
    for (int m = 0; m < 3; ++m) {
      for (int n = 0; n < 3; ++n) {
        float acc = 0.f;
        for (int k = 0; k < 3; ++k) acc += Aw[p][m*3+k] * Rl[k*3+n];
        Aw[j][m*3+n] = acc;
      }
      float tacc = tw[p][m];
      for (int k = 0; k < 3; ++k) tacc += Aw[p][m*3+k] * tl[k];
      tw[j][m] = tacc;
    }
  }
  for (int j = 0; j < NJ; ++j) {
    for (int m = 0; m < 3; ++m) {
      float corr = 0.f;
      for (int n = 0; n < 3; ++n) corr += Aw[j][m*3+n] * Jl[j][n];
      float* g = wsG + ((size_t)b*NJ + j)*12 + m*4;
      g[0] = Aw[j][m*3+0]; g[1] = Aw[j][m*3+1]; g[2] = Aw[j][m*3+2];
      g[3] = tw[j][m] - corr;
    }
  }
}

// ---------------- K3: fused blendshape WMMA GEMM (32v x 32b tile) + skinning ----------------
#define GSTR 289   // padded per-body G stride (gcd(289,64)=1 -> conflict-free)
#define WSTR 25    // padded per-vertex weight stride

__global__ __launch_bounds__(32) void k_lbs_wmma(
    const float* __restrict__ Apack, const float* __restrict__ Bpad,
    const float* __restrict__ v_template, const float* __restrict__ weights,
    const float* __restrict__ G12, float* __restrict__ out) {
  __shared__ float s_vp[96 * 33];        // posed vertex tile [96 rows][32 cols +1 pad]
  __shared__ float s_G[TILE_B * GSTR];   // 32 bodies x (24 joints x 12, padded)
  __shared__ float s_w[TILE_V * WSTR];   // 32 vertices x (24 weights, padded)

  const int lane = threadIdx.x;
  const int vt = blockIdx.x;             // vertex tile (32 vertices -> 96 M rows)
  const int bt = blockIdx.y;             // batch tile (32 bodies)

  for (int i = lane; i < TILE_B * 288; i += 32) {
    int bl = i / 288, e = i - bl * 288;
    s_G[bl * GSTR + e] = G12[((size_t)(bt*TILE_B + bl))*288 + e];
  }
  for (int i = lane; i < TILE_V * 24; i += 32) {
    int vl = i / 24, e = i - vl * 24, v = vt*TILE_V + vl;
    s_w[vl * WSTR + e] = (v < NV) ? weights[(size_t)v*NJ + e] : 0.f;
  }

  // ---- GEMM: C[96 x 32] = A[96 x 220] * B[220 x 32], branch-free inner loop ----
  v8f accA[6] = {}, accB[6] = {};        // [M-block][N-half]
  const int m  = lane & 15;
  const int kh = (lane >> 4) * 2;
  const float* Arow = Apack + (size_t)(vt*96 + m) * KTOT;
  const float* Br0  = Bpad  + (size_t)(bt*TILE_B + m)      * KTOT;
  const float* Br1  = Bpad  + (size_t)(bt*TILE_B + 16 + m) * KTOT;

  for (int k0 = 0; k0 < KTOT; k0 += 4) {
    const int ka = k0 + kh;                       // even -> 8B-aligned b64 loads
    v2f bf0 = *(const v2f*)(Br0 + ka);
    v2f bf1 = *(const v2f*)(Br1 + ka);
#pragma unroll
    for (int q = 0; q < 6; ++q) {
      v2f a = *(const v2f*)(Arow + (size_t)q*16*KTOT + ka);
      accA[q] = __builtin_amdgcn_wmma_f32_16x16x4_f32(false, a, false, bf0, (short)0, accA[q], false, false);
      accB[q] = __builtin_amdgcn_wmma_f32_16x16x4_f32(false, a, false, bf1, (short)0, accB[q], false, false);
    }
  }

  // add template (clamped index, junk rows never stored) and park in LDS
  const int mhalf = (lane >> 4) * 8;
  const int ncol  = lane & 15;
#pragma unroll
  for (int q = 0; q < 6; ++q) {
#pragma unroll
    for (int i = 0; i < 8; ++i) {
      int mrow = q*16 + i + mhalf;
      int Ra = vt*96 + mrow;
      float tv = v_template[Ra < M_ROWS ? Ra : 0];
      s_vp[mrow*33 + ncol     ] = accA[q][i] + tv;
      s_vp[mrow*33 + ncol + 16] = accB[q][i] + tv;
    }
  }
  __syncthreads();

  // ---- skinning: lane = vertex (coalesced stores), iteration = body ----
  for (int it = 0; it < TILE_B; ++it) {
    int vl = lane, bl = it;
    float vx = s_vp[(vl*3 + 0)*33 + bl];
    float vy = s_vp[(vl*3 + 1)*33 + bl];
    float vz = s_vp[(vl*3 + 2)*33 + bl];
    float ox = 0.f, oy = 0.f, oz = 0.f;
    const float* gb = &s_G[bl * GSTR];
#pragma unroll
    for (int j = 0; j < NJ; ++j) {
      float w = s_w[vl*WSTR + j];          // lane-stride 25: conflict-free
      const float* g = gb + j*12;          // wave-uniform: broadcast
      ox += w * (g[0]*vx + g[1]*vy + g[2] *vz + g[3]);
      oy += w * (g[4]*vx + g[5]*vy + g[6] *vz + g[7]);
      oz += w * (g[8]*vx + g[9]*vy + g[10]*vz + g[11]);
    }
    int v = vt*TILE_V + vl, b = bt*TILE_B + bl;
    if (v < NV) {                          // lanes write contiguous 12B triples
      float* o = out + ((size_t)b*NV + v)*3;
      o[0] = ox; o[1] = oy; o[2] = oz;
    }
  }
}

// ---------------- launch ----------------
extern "C" void kernel_launch(void* const* d_in, const int* in_sizes, int n_in,
                              void* d_out, int out_size, void* d_ws, size_t ws_size,
                              hipStream_t stream) {
  const float* pose      = (const float*)d_in[0];
  const float* beta      = (const float*)d_in[1];
  const float* shapedirs = (const float*)d_in[2];
  const float* posedirs  = (const float*)d_in[3];
  const float* v_tmpl    = (const float*)d_in[4];
  const float* Jreg      = (const float*)d_in[5];
  const float* weights   = (const float*)d_in[6];
  float* out = (float*)d_out;
  float* ws  = (float*)d_ws;

  float* wsJSJT = ws + WS_JSJT;
  float* wsB    = ws + WS_B;
  float* wsR    = ws + WS_R;
  float* wsJ    = ws + WS_J;
  float* wsG    = ws + WS_G;
  float* wsA    = ws + WS_A;

  k_pack_A<<<(M_PAD*KTOT + 255)/256, 256, 0, stream>>>(posedirs, shapedirs, wsA);
  k_fold_regressor<<<NJ, 128, 0, stream>>>(Jreg, shapedirs, v_tmpl, wsJSJT);
  k_pose_prep<<<(NB*NJ + 255)/256, 256, 0, stream>>>(pose, beta, wsJSJT, wsR, wsJ, wsB);
  k_fk<<<(NB + 63)/64, 64, 0, stream>>>(wsR, wsJ, wsG);
  dim3 grid(N_VT, NB/TILE_B);  // 216 x 32 tiles
  k_lbs_wmma<<<grid, 32, 0, stream>>>(wsA, wsB, v_tmpl, weights, wsG, out);
}